// HMGCN_43207370997829
// MI455X (gfx1250) — compile-verified
//
#include <hip/hip_runtime.h>
#include <hip/hip_bf16.h>
#include <math.h>

typedef __attribute__((ext_vector_type(2))) float v2f;
typedef __attribute__((ext_vector_type(8))) float v8f;

#define NN 100000
#define EE 1600000
#define FIN 256
#define FOUT 64

// ---------------------------------------------------------------------------
// Degree kernels (deg stored as float; self-loop contributes 1.0)
// ---------------------------------------------------------------------------
__global__ __launch_bounds__(256) void hm_deg_init(float* deg, int n) {
    int i = blockIdx.x * 256 + threadIdx.x;
    if (i < n) deg[i] = 1.0f;
}

__global__ __launch_bounds__(256) void hm_deg_count(const int* __restrict__ dst,
                                                    float* __restrict__ deg) {
    int e = blockIdx.x * 256 + threadIdx.x;
    if (e < EE) atomicAdd(&deg[dst[e]], 1.0f);
}

__global__ __launch_bounds__(256) void hm_deg_rsqrt(float* deg, int n) {
    int i = blockIdx.x * 256 + threadIdx.x;
    if (i < n) deg[i] = rsqrtf(deg[i]);   // deg >= 1 always (self-loop)
}

__global__ __launch_bounds__(256) void hm_zero(float* p, int n) {
    int i = blockIdx.x * 256 + threadIdx.x;
    if (i < n) p[i] = 0.0f;
}

// ---------------------------------------------------------------------------
// GEMM: H = feature @ W   ([NN x 256] x [256 x 64]) via V_WMMA_F32_16X16X4_F32
// One wave -> 16 rows x 64 cols; 8 waves per block -> 128 rows per block.
// W staged in 64KB LDS in k-pair-interleaved layout so each lane's B fragment
// {W[k][c], W[k+1][c]} is one contiguous 8-byte ds_load_b64.
//   Wl[(k>>1)*128 + c*2 + (k&1)] = W[k*64 + c]
// ---------------------------------------------------------------------------
__global__ __launch_bounds__(256)
void hm_gemm_wmma(const float* __restrict__ F, const float* __restrict__ W,
                  float* __restrict__ H) {
    __shared__ float Wl[FIN * FOUT];          // 64 KB
    int tid = threadIdx.x;
    for (int i = tid; i < FIN * FOUT; i += 256) {
        int k = i >> 6;                       // 0..255
        int c = i & 63;
        Wl[(k >> 1) * 128 + c * 2 + (k & 1)] = W[i];
    }
    __syncthreads();

    int wave = tid >> 5;
    int lane = tid & 31;
    int tile = blockIdx.x * 8 + wave;         // 16-row tile index
    if (tile * 16 >= NN) return;              // wave-uniform exit (EXEC stays full)
    int row0 = tile * 16;

    int lm   = lane & 15;                     // row within tile (A) / col within tile (B,C)
    int koff = (lane >> 4) << 1;              // 0 for lanes 0-15, 2 for lanes 16-31

    const float* Arow = F + (size_t)(row0 + lm) * FIN;

    v8f acc0 = {}, acc1 = {}, acc2 = {}, acc3 = {};

#pragma unroll 8
    for (int k = 0; k < FIN; k += 4) {
        // A fragment 16x4: lane holds A[lm][k+koff .. k+koff+1] (contiguous, 8B aligned)
        v2f a = *(const v2f*)(Arow + k + koff);
        // B fragments: lane holds {W[k+koff][c], W[k+koff+1][c]} as one b64 load
        const float* Wp = &Wl[((k + koff) >> 1) * 128 + lm * 2];
        v2f b0 = *(const v2f*)(Wp);           // cols  0..15
        v2f b1 = *(const v2f*)(Wp + 32);      // cols 16..31
        v2f b2 = *(const v2f*)(Wp + 64);      // cols 32..47
        v2f b3 = *(const v2f*)(Wp + 96);      // cols 48..63
        acc0 = __builtin_amdgcn_wmma_f32_16x16x4_f32(false, a, false, b0, (short)0, acc0, false, false);
        acc1 = __builtin_amdgcn_wmma_f32_16x16x4_f32(false, a, false, b1, (short)0, acc1, false, false);
        acc2 = __builtin_amdgcn_wmma_f32_16x16x4_f32(false, a, false, b2, (short)0, acc2, false, false);
        acc3 = __builtin_amdgcn_wmma_f32_16x16x4_f32(false, a, false, b3, (short)0, acc3, false, false);
    }

    // C/D layout: VGPR v, lanes 0-15 -> M=v, lanes 16-31 -> M=v+8; N = lm
    int mhalf = (lane >> 4) * 8;
    float* Hrow = H + (size_t)row0 * FOUT;
#pragma unroll
    for (int v = 0; v < 8; ++v) {
        int m = v + mhalf;
        Hrow[(size_t)m * FOUT + lm]      = acc0[v];
        Hrow[(size_t)m * FOUT + 16 + lm] = acc1[v];
        Hrow[(size_t)m * FOUT + 32 + lm] = acc2[v];
        Hrow[(size_t)m * FOUT + 48 + lm] = acc3[v];
    }
}

// ---------------------------------------------------------------------------
// Self-loop init: X[n][f] = H[n][f] * dinv[n]^2
// ---------------------------------------------------------------------------
__global__ __launch_bounds__(256)
void hm_selfloop(const float* __restrict__ H, const float* __restrict__ dinv,
                 float* __restrict__ X) {
    size_t i = (size_t)blockIdx.x * 256 + threadIdx.x;
    if (i >= (size_t)NN * FOUT) return;
    int n = (int)(i >> 6);
    float di = dinv[n];
    X[i] = H[i] * di * di;
}

// ---------------------------------------------------------------------------
// Edge aggregation: 16 threads per edge, float4 each; L2-resident atomics.
// ---------------------------------------------------------------------------
__global__ __launch_bounds__(256)
void hm_edge_agg(const int* __restrict__ src, const int* __restrict__ dst,
                 const float* __restrict__ H, const float* __restrict__ dinv,
                 float* __restrict__ X) {
    size_t tid = (size_t)blockIdx.x * 256 + threadIdx.x;
    if (tid >= (size_t)EE * 16) return;
    int e = (int)(tid >> 4);
    int q = (int)(tid & 15) * 4;
    int s = src[e], d = dst[e];
    float norm = dinv[s] * dinv[d];
    const float4 h = *(const float4*)&H[(size_t)s * FOUT + q];
    float* xp = &X[(size_t)d * FOUT + q];
    atomicAdd(xp + 0, h.x * norm);
    atomicAdd(xp + 1, h.y * norm);
    atomicAdd(xp + 2, h.z * norm);
    atomicAdd(xp + 3, h.w * norm);
}

// ---------------------------------------------------------------------------
// Epilogue: x = relu(X + b) in place, accumulate column sums (4 nodes/block)
// ---------------------------------------------------------------------------
__global__ __launch_bounds__(256)
void hm_relu_colsum(float* __restrict__ X, const float* __restrict__ bias,
                    float* __restrict__ colsum) {
    __shared__ float lds[256];
    int tid = threadIdx.x;
    int f = tid & 63;
    size_t idx = (size_t)blockIdx.x * 256 + tid;   // == n*64+f ; NN*64 % 256 == 0
    float v = fmaxf(X[idx] + bias[f], 0.0f);
    X[idx] = v;
    lds[tid] = v;
    __syncthreads();
    if (tid < 64) {
        float s = lds[tid] + lds[tid + 64] + lds[tid + 128] + lds[tid + 192];
        atomicAdd(&colsum[tid], s);
    }
}

// ---------------------------------------------------------------------------
// Attention: hm = sigmoid(colsum/N); s0/s1 = hm·w ; beta = softmax over 4 rel.
// One block, 64 threads.
// ---------------------------------------------------------------------------
__global__ __launch_bounds__(64)
void hm_attention(const float* __restrict__ colsum, const float* __restrict__ w1,
                  const float* __restrict__ w2, float* __restrict__ beta) {
    __shared__ float red[64];
    __shared__ float s0[4], s1[4];
    int f = threadIdx.x;
    for (int r = 0; r < 4; ++r) {
        float hm = colsum[r * 64 + f] * (1.0f / (float)NN);
        hm = 1.0f / (1.0f + expf(-hm));
        red[f] = hm * w1[f];
        __syncthreads();
        for (int off = 32; off >= 1; off >>= 1) {
            if (f < off) red[f] += red[f + off];
            __syncthreads();
        }
        if (f == 0) s0[r] = red[0];
        __syncthreads();
        red[f] = hm * w2[f];
        __syncthreads();
        for (int off = 32; off >= 1; off >>= 1) {
            if (f < off) red[f] += red[f + off];
            __syncthreads();
        }
        if (f == 0) s1[r] = red[0];
        __syncthreads();
    }
    if (f == 0) {
        float m0 = s0[0], m1 = s1[0];
        for (int r = 1; r < 4; ++r) { m0 = fmaxf(m0, s0[r]); m1 = fmaxf(m1, s1[r]); }
        float e0[4], e1[4], t0 = 0.f, t1 = 0.f;
        for (int r = 0; r < 4; ++r) {
            e0[r] = expf(s0[r] - m0); t0 += e0[r];
            e1[r] = expf(s1[r] - m1); t1 += e1[r];
        }
        for (int r = 0; r < 4; ++r) {
            beta[r]     = e0[r] / t0;   // type-0 weights
            beta[4 + r] = e1[r] / t1;   // type-1 weights
        }
    }
}

// ---------------------------------------------------------------------------
// Stable partition scan: rowmap[n] = output row (type-0 rows first, in order)
// One block, 1024 threads, Hillis-Steele scan over per-thread chunk counts.
// ---------------------------------------------------------------------------
__global__ __launch_bounds__(1024)
void hm_scan_types(const int* __restrict__ types, int* __restrict__ rowmap) {
    __shared__ int cnt[1024];
    int tid = threadIdx.x;
    const int chunk = (NN + 1023) / 1024;
    int start = tid * chunk;
    int end = start + chunk; if (end > NN) end = NN; if (start > NN) start = NN;
    int c0 = 0;
    for (int n = start; n < end; ++n) c0 += (types[n] == 0);
    cnt[tid] = c0;
    __syncthreads();
    for (int off = 1; off < 1024; off <<= 1) {
        int v = cnt[tid];
        int add = (tid >= off) ? cnt[tid - off] : 0;
        __syncthreads();
        cnt[tid] = v + add;
        __syncthreads();
    }
    int total0 = cnt[1023];
    int pre0 = cnt[tid] - c0;          // type-0 nodes before this chunk
    int p0 = pre0;
    int p1 = start - pre0;             // type-1 nodes before this chunk
    for (int n = start; n < end; ++n) {
        if (types[n] == 0) rowmap[n] = p0++;
        else               rowmap[n] = total0 + p1++;
    }
}

// ---------------------------------------------------------------------------
// Final combine: out[rowmap[n]] = sum_r beta[type*4+r] * X_r[n]
// ---------------------------------------------------------------------------
__global__ __launch_bounds__(256)
void hm_combine(const float* __restrict__ X, const int* __restrict__ types,
                const int* __restrict__ rowmap, const float* __restrict__ beta,
                float* __restrict__ out) {
    size_t tid = (size_t)blockIdx.x * 256 + threadIdx.x;
    if (tid >= (size_t)NN * 16) return;
    int n = (int)(tid >> 4);
    int q = (int)(tid & 15) * 4;
    const float* bp = beta + types[n] * 4;
    float4 acc = {0.f, 0.f, 0.f, 0.f};
#pragma unroll
    for (int r = 0; r < 4; ++r) {
        const float4 h = *(const float4*)&X[((size_t)r * NN + n) * FOUT + q];
        float w = bp[r];
        acc.x += w * h.x; acc.y += w * h.y; acc.z += w * h.z; acc.w += w * h.w;
    }
    *(float4*)&out[(size_t)rowmap[n] * FOUT + q] = acc;
}

// ---------------------------------------------------------------------------
extern "C" void kernel_launch(void* const* d_in, const int* in_sizes, int n_in,
                              void* d_out, int out_size, void* d_ws, size_t ws_size,
                              hipStream_t stream) {
    const float* feature = (const float*)d_in[0];
    const int* ei[4] = {(const int*)d_in[1], (const int*)d_in[2],
                        (const int*)d_in[3], (const int*)d_in[4]};
    const int* types = (const int*)d_in[5];
    const float* W[4] = {(const float*)d_in[6], (const float*)d_in[8],
                         (const float*)d_in[10], (const float*)d_in[12]};
    const float* bia[4] = {(const float*)d_in[7], (const float*)d_in[9],
                           (const float*)d_in[11], (const float*)d_in[13]};
    const float* w1 = (const float*)d_in[14];
    const float* w2 = (const float*)d_in[15];
    float* out = (float*)d_out;

    // workspace layout (floats)
    float* X      = (float*)d_ws;                       // 4*NN*64
    float* Htmp   = X + (size_t)4 * NN * FOUT;          // NN*64
    float* dinv   = Htmp + (size_t)NN * FOUT;           // 4*NN (also used as deg)
    float* colsum = dinv + (size_t)4 * NN;              // 4*64
    float* beta   = colsum + 4 * FOUT;                  // 8
    int*   rowmap = (int*)(beta + 8);                   // NN

    // degrees -> dinv
    hm_deg_init<<<(4 * NN + 255) / 256, 256, 0, stream>>>(dinv, 4 * NN);
    for (int r = 0; r < 4; ++r)
        hm_deg_count<<<(EE + 255) / 256, 256, 0, stream>>>(ei[r] + EE, dinv + (size_t)r * NN);
    hm_deg_rsqrt<<<(4 * NN + 255) / 256, 256, 0, stream>>>(dinv, 4 * NN);

    hm_zero<<<1, 256, 0, stream>>>(colsum, 4 * FOUT);

    const int gemm_blocks = ((NN / 16) + 7) / 8;        // 782
    for (int r = 0; r < 4; ++r) {
        float* Xr = X + (size_t)r * NN * FOUT;
        const float* dr = dinv + (size_t)r * NN;
        hm_gemm_wmma<<<gemm_blocks, 256, 0, stream>>>(feature, W[r], Htmp);
        hm_selfloop<<<(NN * FOUT + 255) / 256, 256, 0, stream>>>(Htmp, dr, Xr);
        hm_edge_agg<<<(int)(((size_t)EE * 16 + 255) / 256), 256, 0, stream>>>(
            ei[r], ei[r] + EE, Htmp, dr, Xr);
        hm_relu_colsum<<<(NN * FOUT) / 256, 256, 0, stream>>>(Xr, bia[r], colsum + r * FOUT);
    }

    hm_attention<<<1, 64, 0, stream>>>(colsum, w1, w2, beta);
    hm_scan_types<<<1, 1024, 0, stream>>>(types, rowmap);
    hm_combine<<<(int)(((size_t)NN * 16 + 255) / 256), 256, 0, stream>>>(
        X, types, rowmap, beta, out);
}